// AssignedDeepPolyConvTransformer_17557826306519
// MI455X (gfx1250) — compile-verified
//
#include <hip/hip_runtime.h>

typedef float v2f __attribute__((ext_vector_type(2)));
typedef float v4f __attribute__((ext_vector_type(4)));
typedef float v8f __attribute__((ext_vector_type(8)));

#define C_IN    8
#define C_OUT   32
#define HO      16
#define WO      16
#define IN_FEAT 8192
#define OUT_FEAT 8192
#define KDIM    128   // C_IN * 4 * 4

// ---------------------------------------------------------------------------
// Kernel 1 (dominant, memory-bound): fill the 8192x8192 Toeplitz matrix.
// One block per row; float4 non-temporal streaming stores (256 MB > 192 MB L2,
// so bypass); fast-path zero quads; rare nonzero quads gather from the 16 KB
// weight tensor (L2-resident). Also writes bias_backsub[row].
// ---------------------------------------------------------------------------
__global__ void toeplitz_fill_kernel(const float* __restrict__ weight,
                                     const float* __restrict__ bias,
                                     float* __restrict__ wmat,
                                     float* __restrict__ bias_backsub) {
  const int row   = blockIdx.x;       // 0..8191  = (c_out, ho, wo)
  const int c_out = row >> 8;
  const int p     = row & 255;
  const int h0    = 2 * (p >> 4) - 1; // padding P=1, stride S=2
  const int w0    = 2 * (p & 15) - 1;
  const float* wrow = weight + c_out * KDIM;
  float* out = wmat + (size_t)row * IN_FEAT;

  if (threadIdx.x == 0) bias_backsub[row] = bias[c_out];

#pragma unroll
  for (int it = 0; it < 8; ++it) {
    const int f    = it * 256 + threadIdx.x;   // float4 index within row, 0..2047
    const int c_in = f >> 8;
    const int h    = (f >> 3) & 31;
    const int wb   = (f & 7) << 2;             // aligned w base: 0,4,...,28
    const int kh   = h - h0;
    v4f val = {0.f, 0.f, 0.f, 0.f};
    if ((unsigned)kh < 4u && (wb + 3 >= w0) && (wb <= w0 + 3)) {
#pragma unroll
      for (int j = 0; j < 4; ++j) {
        const int kw = wb + j - w0;
        if ((unsigned)kw < 4u)
          val[j] = wrow[c_in * 16 + kh * 4 + kw];
      }
    }
    __builtin_nontemporal_store(val, (v4f*)(out + 4 * f));
  }
}

// ---------------------------------------------------------------------------
// Kernel 2: interval bound propagation as f32 WMMA GEMMs.
// out[bound] tile(16x16) = sum_k  Wp(16x4) x B_srcP(4x16)  +  Wm x B_srcM.
// im2col of the bound tensors is built on the fly per the ISA lane layout.
// One wave (32 threads) per (bound, mtile, ntile); 64 waves total.
// OOB taps use clamped-index unconditional loads + cndmask (no EXEC churn).
// ---------------------------------------------------------------------------
__global__ void bounds_wmma_kernel(const float* __restrict__ bounds,
                                   const float* __restrict__ weight,
                                   const float* __restrict__ bias,
                                   float* __restrict__ out_bounds) {
  const int lane  = threadIdx.x;        // 0..31, EXEC all ones
  const int bi    = blockIdx.x;         // 0..63
  const int bound = bi >> 5;            // 0 = lower, 1 = upper
  const int mtile = (bi >> 4) & 1;      // c_out half
  const int ntile = bi & 15;            // 16-position chunk
  const int cbase = mtile * 16;
  const int hi    = lane >> 4;          // lane half selects K{0,1} vs K{2,3}
  const int m     = lane & 15;          // A-matrix row (M) for this lane
  const int n     = lane & 15;          // B/C/D column (N) for this lane
  const int p     = ntile * 16 + n;     // output spatial index
  const int ho    = p >> 4, wo = p & 15;

  // lower: Wp applied to l (bounds[0]), Wm applied to u (bounds[1]); upper: swapped
  const float* srcP = bounds + bound * IN_FEAT;
  const float* srcM = bounds + (1 - bound) * IN_FEAT;

  v8f acc = {0.f, 0.f, 0.f, 0.f, 0.f, 0.f, 0.f, 0.f};

  for (int k0 = 0; k0 < KDIM; k0 += 4) {
    // A fragment: 16x4 f32, lanes 0-15 hold K = k0+{0,1}, lanes 16-31 K = k0+{2,3}
    const int kA = k0 + 2 * hi;
    const float wv0 = weight[(cbase + m) * KDIM + kA];     // merges to b64
    const float wv1 = weight[(cbase + m) * KDIM + kA + 1];
    v2f ap = { fmaxf(wv0, 0.f), fmaxf(wv1, 0.f) };
    v2f am = { fminf(wv0, 0.f), fminf(wv1, 0.f) };

    // B fragment: 4x16 f32, VGPR0 rows K = k0 + 2*hi, VGPR1 rows K = k0+2*hi+1,
    // N = lane&15. im2col (stride 2, pad 1); zero-pad via clamped load + select.
    v2f bp, bm;
#pragma unroll
    for (int q = 0; q < 2; ++q) {
      const int r    = k0 + 2 * hi + q;       // reduction index = (c_in, kh, kw)
      const int c_in = r >> 4;
      const int kh   = (r >> 2) & 3;
      const int kw   = r & 3;
      const int hpos = 2 * ho - 1 + kh;
      const int wpos = 2 * wo - 1 + kw;
      const bool ok  = ((unsigned)hpos < 32u) & ((unsigned)wpos < 32u);
      const int idx  = ok ? (c_in * 1024 + hpos * 32 + wpos) : 0;  // always valid
      const float vp = srcP[idx];             // unconditional, EXEC untouched
      const float vm = srcM[idx];
      bp[q] = ok ? vp : 0.f;                  // v_cndmask
      bm[q] = ok ? vm : 0.f;
    }

    acc = __builtin_amdgcn_wmma_f32_16x16x4_f32(false, ap, false, bp,
                                                (short)0, acc, false, false);
    acc = __builtin_amdgcn_wmma_f32_16x16x4_f32(false, am, false, bm,
                                                (short)0, acc, false, false);
  }

  // C/D layout: VGPR v, lanes 0-15 -> M = v, lanes 16-31 -> M = v + 8
#pragma unroll
  for (int v = 0; v < 8; ++v) {
    const int co = cbase + v + hi * 8;
    out_bounds[bound * OUT_FEAT + co * 256 + p] = acc[v] + bias[co];
  }
}

// ---------------------------------------------------------------------------
extern "C" void kernel_launch(void* const* d_in, const int* in_sizes, int n_in,
                              void* d_out, int out_size, void* d_ws, size_t ws_size,
                              hipStream_t stream) {
  const float* bounds = (const float*)d_in[0];   // [2,8,32,32]
  const float* weight = (const float*)d_in[1];   // [32,8,4,4]
  const float* bias   = (const float*)d_in[2];   // [32]
  // d_in[3] = assignment (unused by the reference forward)

  float* out          = (float*)d_out;
  float* out_bounds   = out;                                   // 16384 floats
  float* wmat         = out + 2 * OUT_FEAT;                    // 8192*8192 floats
  float* bias_backsub = wmat + (size_t)OUT_FEAT * IN_FEAT;     // 8192 floats

  bounds_wmma_kernel<<<64, 32, 0, stream>>>(bounds, weight, bias, out_bounds);
  toeplitz_fill_kernel<<<OUT_FEAT, 256, 0, stream>>>(weight, bias, wmat, bias_backsub);
}